// PixelShuffle_v3_35407710388832
// MI455X (gfx1250) — compile-verified
//
#include <hip/hip_runtime.h>

typedef float v2f __attribute__((ext_vector_type(2)));
typedef float v8f __attribute__((ext_vector_type(8)));

#define CIN   128
#define COUT  512
#define HIN   200
#define WIN   200
#define HW    (HIN * WIN)        /* 40000 spatial positions per batch      */
#define NBLK  64                 /* N columns per workgroup                */
#define NTILES (HW / NBLK)       /* 625 N-blocks per batch                 */
#define PPITCH 160               /* dwords per K-row-pair: 128 + 32 skew   */
#define EPSV  1e-5f

__global__ __launch_bounds__(256)
void conv1x1_bn_relu_ps_kernel(const float* __restrict__ x,
                               const float* __restrict__ Wm,
                               const float* __restrict__ gamma,
                               const float* __restrict__ beta,
                               const float* __restrict__ rmean,
                               const float* __restrict__ rvar,
                               float* __restrict__ out)
{
    // x tile stored pre-swizzled into WMMA-B fragment order:
    //   element (c, n) -> xs[(c>>1)*PPITCH + 2*n + (c&1)]
    // so a lane's (K, K+1) pair is one contiguous, 8B-aligned dword pair.
    __shared__ float xs[(CIN / 2) * PPITCH];   // 64 * 160 * 4 = 40 KB

    const int tid  = threadIdx.x;
    const int lane = tid & 31;                 // wave32
    const int wave = tid >> 5;                 // 8 waves per block
    const int m    = lane & 15;
    const int half = lane >> 4;                // selects K offset +2 / M offset +8

    // ---- block decomposition: (batch, M-block of 128, N-block of 64) ----
    int blk = blockIdx.x;
    const int nb = blk % NTILES;  blk /= NTILES;
    const int mb = blk & 3;
    const int b  = blk >> 2;
    const int n0     = nb * NBLK;
    const int o_base = mb * 128 + wave * 16;   // this wave's 16 output channels

    // ---- stage x tile [CIN x NBLK] -> LDS, interleaving row pairs ----------
    const float* xb = x + (size_t)b * CIN * HW + n0;
    #pragma unroll
    for (int it = 0; it < 4; ++it) {
        const int u  = it * 256 + tid;         // 1024 units: (row pair) x (col quad)
        const int p  = u >> 4;                 // row pair   0..63  (rows 2p, 2p+1)
        const int qq = u & 15;                 // col quad   0..15  (cols 4qq..4qq+3)
        const float* s0 = xb + (size_t)(2 * p) * HW + 4 * qq;
        const float* s1 = s0 + HW;
        const float4 va = *(const float4*)s0;  // row 2p
        const float4 vb = *(const float4*)s1;  // row 2p+1
        __builtin_prefetch(s0 + NBLK, 0, 0);   // warm L2 for next N-block (global_prefetch_b8)
        float* dst = &xs[p * PPITCH + 8 * qq];
        const float4 w0 = {va.x, vb.x, va.y, vb.y};
        const float4 w1 = {va.z, vb.z, va.w, vb.w};
        *(float4*)(dst)     = w0;              // ds_store_b128
        *(float4*)(dst + 4) = w1;
    }

    // ---- A fragments: 16 rows of W, full K=128, kept in registers ----------
    // ISA layout (f32 16x4 A): VGPR0 <- K=4j+2*half, VGPR1 <- K+1
    v2f a[32];
    {
        const float* wr = Wm + (size_t)(o_base + m) * CIN + 2 * half;
        #pragma unroll
        for (int j = 0; j < 32; ++j)
            a[j] = *(const v2f*)(wr + 4 * j);
    }

    // ---- per-lane BN scale/shift for the 8 M-rows this lane holds ----------
    float scl[8], sft[8];
    #pragma unroll
    for (int v = 0; v < 8; ++v) {
        const int o = o_base + v + 8 * half;   // C/D layout: M = v + 8*half
        const float inv = rsqrtf(rvar[o] + EPSV);
        const float s   = gamma[o] * inv;
        scl[v] = s;
        sft[v] = beta[o] - rmean[o] * s;
    }

    __syncthreads();

    // ---- GEMM: 4 N-subtiles of 16, K = 128 in steps of 4 -------------------
    // One base VGPR; every (j, t) LDS offset is a compile-time immediate.
    const float* bp = &xs[half * PPITCH + 2 * m];

    v8f acc[4];
    #pragma unroll
    for (int t = 0; t < 4; ++t) acc[t] = 0.0f;

    #pragma unroll
    for (int j = 0; j < 32; ++j) {
        #pragma unroll
        for (int t = 0; t < 4; ++t) {
            const v2f bf = *(const v2f*)(bp + j * (2 * PPITCH) + t * 32); // ds_load_b64
            acc[t] = __builtin_amdgcn_wmma_f32_16x16x4_f32(
                false, a[j], false, bf, (short)0, acc[t], false, false);
        }
    }

    // ---- BN + ReLU + pixel-shuffle scatter ---------------------------------
    // o = 256*i + 128*jj + c2  ->  out[b, c2, 2h+jj, 2w+i]; o_base is a
    // multiple of 16, so the i/jj bits are wave-uniform.
    const int jbit = (o_base >> 7) & 1;
    const int ibit = (o_base >> 8) & 1;
    const int c2b  = (o_base & 127) + 8 * half;

    #pragma unroll
    for (int t = 0; t < 4; ++t) {
        const int n  = n0 + t * 16 + m;        // spatial index, N = lane&15
        const int hh = n / WIN;
        const int ww = n - hh * WIN;
        const size_t sp = (size_t)(2 * hh + jbit) * (2 * WIN) + (size_t)(2 * ww + ibit);
        float* obase = out + ((size_t)b * 128 + c2b) * (size_t)(4 * HW) + sp;
        #pragma unroll
        for (int v = 0; v < 8; ++v) {
            const float y = fmaxf(fmaf(acc[t][v], scl[v], sft[v]), 0.0f);
            obase[(size_t)v * (size_t)(4 * HW)] = y;
        }
    }
}

extern "C" void kernel_launch(void* const* d_in, const int* in_sizes, int n_in,
                              void* d_out, int out_size, void* d_ws, size_t ws_size,
                              hipStream_t stream) {
    const float* x     = (const float*)d_in[0];
    const float* Wm    = (const float*)d_in[1];
    const float* gamma = (const float*)d_in[2];
    const float* beta  = (const float*)d_in[3];
    const float* rmean = (const float*)d_in[4];
    const float* rvar  = (const float*)d_in[5];
    float* out = (float*)d_out;

    const int grid = 4 /*batch*/ * 4 /*M-blocks*/ * NTILES;   // 10000 blocks
    conv1x1_bn_relu_ps_kernel<<<grid, 256, 0, stream>>>(
        x, Wm, gamma, beta, rmean, rvar, out);
}